// MPNN_79044578115931
// MI455X (gfx1250) — compile-verified
//
#include <hip/hip_runtime.h>
#include <hip/hip_bf16.h>

#define N_NODES 50000
#define N_EDGES 320000
#define HIDDEN  256
#define KDIM    512       // 2*HIDDEN (concat of h and agg)
#define N_LAYERS 5
#define BN_EPS  1e-5f
#define TK      64        // K-chunk staged in LDS
#define BPS     80        // sBp kp-row stride (floats-pairs): 2*BPS % 64 == 32 -> half-waves on disjoint banks

typedef __attribute__((ext_vector_type(2))) float v2f;
typedef __attribute__((ext_vector_type(8))) float v8f;

// ---------------------------------------------------------------- degree / norm
__global__ void deg_kernel(const int* __restrict__ col, float* __restrict__ deg) {
    int e = blockIdx.x * blockDim.x + threadIdx.x;
    if (e < N_EDGES) atomicAdd(&deg[col[e]], 1.0f);
}

__global__ void dinv_kernel(float* __restrict__ deg) {
    int i = blockIdx.x * blockDim.x + threadIdx.x;
    if (i < N_NODES) {
        float d = deg[i];
        deg[i] = (d > 0.0f) ? rsqrtf(d) : 0.0f;
    }
}

__global__ void norm_kernel(const int* __restrict__ row, const int* __restrict__ col,
                            const float* __restrict__ dinv, float* __restrict__ nrm) {
    int e = blockIdx.x * blockDim.x + threadIdx.x;
    if (e < N_EDGES) nrm[e] = dinv[row[e]] * dinv[col[e]];
}

// -------------------------------------------------- message + scatter-add (per edge)
// block = 256 threads = one edge; thread t = feature t
__global__ void msg_scatter_kernel(const float* __restrict__ h,
                                   const float* __restrict__ ea,   // [E][4]
                                   const int*   __restrict__ row,
                                   const int*   __restrict__ col,
                                   const float* __restrict__ nrm,
                                   const float* __restrict__ ew1,  // [128]
                                   const float* __restrict__ ew2,  // [3][128]
                                   float* __restrict__ agg) {
    const int e = blockIdx.x;
    const int f = threadIdx.x;
    const int r = row[e];
    const int c = col[e];
    const float nw = nrm[e];
    float fe;
    if (f < HIDDEN / 2) {
        fe = ea[e * 4 + 3] * ew1[f];
    } else {
        const int j = f - HIDDEN / 2;
        fe = ea[e * 4 + 0] * ew2[j]
           + ea[e * 4 + 1] * ew2[128 + j]
           + ea[e * 4 + 2] * ew2[256 + j];
    }
    const float v = nw * (h[(size_t)r * HIDDEN + f] + fe);
    atomicAdd(&agg[(size_t)c * HIDDEN + f], v);
}

// ---------------------------------------------------------------- WMMA f32 GEMM
// out[m][n] = relu( sum_k X[m][k] * W[k][n] + bias[n] ),  X = concat(h, agg)
// block = 128 threads = 4 waves; block tile = 64(M) x 64(N); wave tile = 16 x 64
// B staged K-pair-interleaved: sBp[kp][n] = {W[2kp][n], W[2kp+1][n]} so each B
// fragment is a single aligned ds_load_b64 into an even VGPR pair (no packing movs).
__global__ __launch_bounds__(128) void gemm_kernel(const float* __restrict__ hin,
                                                   const float* __restrict__ agg,
                                                   const float* __restrict__ W,     // [512][256]
                                                   const float* __restrict__ bias,  // [256]
                                                   float* __restrict__ out) {
    __shared__ float sA[64][TK + 4];                 // stride 68: conflict-free row access
    __shared__ float sBp[(TK / 2) * BPS * 2];        // 32 kp-rows * 80 pairs * 2 = 20 KB

    const int tid  = threadIdx.x;
    const int wave = tid >> 5;
    const int lane = tid & 31;
    const int hv   = lane >> 4;     // half-wave select (K sub-pair / M sub-block)
    const int l16  = lane & 15;
    const int mBase = blockIdx.x * 64;
    const int n0    = blockIdx.y * 64;

    v8f acc0 = {}, acc1 = {}, acc2 = {}, acc3 = {};

    for (int kb = 0; kb < KDIM / TK; ++kb) {
        const float* src = (kb < 4) ? hin : agg;
        const int kOff = (kb < 4) ? kb * TK : (kb - 4) * TK;
        const int rsub = tid >> 4;          // 0..7
        const int c4   = (tid & 15) * 4;    // 0..60

        // stage X: 64 rows x 64 k
#pragma unroll
        for (int p = 0; p < 8; ++p) {
            const int r = rsub + p * 8;     // 0..63
            const int gr = mBase + r;
            float4 va = make_float4(0.f, 0.f, 0.f, 0.f);
            if (gr < N_NODES)
                va = *(const float4*)(src + (size_t)gr * HIDDEN + kOff + c4);
            sA[r][c4 + 0] = va.x; sA[r][c4 + 1] = va.y;
            sA[r][c4 + 2] = va.z; sA[r][c4 + 3] = va.w;
        }
        // stage W: 64 k x 64 n, K-pair interleaved
#pragma unroll
        for (int p = 0; p < 4; ++p) {
            const int kp = rsub + p * 8;    // 0..31
            const float4 w0 = *(const float4*)(W + (size_t)(kb * TK + 2 * kp    ) * HIDDEN + n0 + c4);
            const float4 w1 = *(const float4*)(W + (size_t)(kb * TK + 2 * kp + 1) * HIDDEN + n0 + c4);
            float* dst = sBp + (size_t)(kp * BPS + c4) * 2;   // 32B-aligned
            dst[0] = w0.x; dst[1] = w1.x; dst[2] = w0.y; dst[3] = w1.y;
            dst[4] = w0.z; dst[5] = w1.z; dst[6] = w0.w; dst[7] = w1.w;
        }
        __syncthreads();

        const int aRow = wave * 16 + l16;
#pragma unroll
        for (int k0 = 0; k0 < TK; k0 += 4) {
            const int kk = k0 + 2 * hv;          // lanes 0-15: K={k0,k0+1}; 16-31: {k0+2,k0+3}
            v2f a; a.x = sA[aRow][kk]; a.y = sA[aRow][kk + 1];
            const int kp = (k0 >> 1) + hv;       // pair row per half-wave
            const float* bp = sBp + (size_t)(kp * BPS + l16) * 2;
            const v2f b0 = *(const v2f*)(bp +  0);   // tile n-offsets: 16 cols = 32 floats
            const v2f b1 = *(const v2f*)(bp + 32);
            const v2f b2 = *(const v2f*)(bp + 64);
            const v2f b3 = *(const v2f*)(bp + 96);
            acc0 = __builtin_amdgcn_wmma_f32_16x16x4_f32(false, a, false, b0, (short)0, acc0, false, false);
            acc1 = __builtin_amdgcn_wmma_f32_16x16x4_f32(false, a, false, b1, (short)0, acc1, false, false);
            acc2 = __builtin_amdgcn_wmma_f32_16x16x4_f32(false, a, false, b2, (short)0, acc2, false, false);
            acc3 = __builtin_amdgcn_wmma_f32_16x16x4_f32(false, a, false, b3, (short)0, acc3, false, false);
        }
        __syncthreads();
    }

    // D layout: VGPR r holds M = r + 8*hv, N = l16 (per 16-col tile)
    const int m0 = mBase + wave * 16;
    if (m0 >= N_NODES) return;
#pragma unroll
    for (int r = 0; r < 8; ++r) {
        const int gm = m0 + r + hv * 8;
        float* o = out + (size_t)gm * HIDDEN + n0;
        float v;
        v = acc0[r] + bias[n0 +  0 + l16]; o[ 0 + l16] = v > 0.f ? v : 0.f;
        v = acc1[r] + bias[n0 + 16 + l16]; o[16 + l16] = v > 0.f ? v : 0.f;
        v = acc2[r] + bias[n0 + 32 + l16]; o[32 + l16] = v > 0.f ? v : 0.f;
        v = acc3[r] + bias[n0 + 48 + l16]; o[48 + l16] = v > 0.f ? v : 0.f;
    }
}

// ---------------------------------------------------------------- batch-norm
// block = 256 threads; thread t owns column t; 64-row chunk per block (coalesced)
__global__ void bn_stats_kernel(const float* __restrict__ h, float* __restrict__ sums) {
    const int t = threadIdx.x;
    const int r0 = blockIdx.x * 64;
    float s = 0.f, s2 = 0.f;
#pragma unroll 4
    for (int i = 0; i < 64; ++i) {
        const int r = r0 + i;
        if (r < N_NODES) {
            const float v = h[(size_t)r * HIDDEN + t];
            s += v; s2 += v * v;
        }
    }
    atomicAdd(&sums[t], s);
    atomicAdd(&sums[HIDDEN + t], s2);
}

__global__ void bn_finalize_kernel(const float* __restrict__ sums,
                                   const float* __restrict__ g,
                                   const float* __restrict__ b,
                                   float* __restrict__ ss) {
    const int t = threadIdx.x;
    const float mu  = sums[t] * (1.0f / N_NODES);
    const float var = sums[HIDDEN + t] * (1.0f / N_NODES) - mu * mu;  // biased var
    const float sc  = g[t] * rsqrtf(var + BN_EPS);
    ss[t] = sc;
    ss[HIDDEN + t] = b[t] - mu * sc;
}

__global__ void bn_apply_kernel(const float* __restrict__ ss, float* __restrict__ h, int relu) {
    const size_t i = (size_t)blockIdx.x * blockDim.x + threadIdx.x;  // grid covers N*H exactly
    const int c = (int)(i & (HIDDEN - 1));
    float v = h[i] * ss[c] + ss[HIDDEN + c];
    if (relu) v = v > 0.f ? v : 0.f;
    h[i] = v;
}

// ---------------------------------------------------------------- launch
extern "C" void kernel_launch(void* const* d_in, const int* in_sizes, int n_in,
                              void* d_out, int out_size, void* d_ws, size_t ws_size,
                              hipStream_t stream) {
    const float* x     = (const float*)d_in[0];
    const int*   ei    = (const int*)d_in[1];   // int32 on device (JAX x64 disabled)
    const float* ea    = (const float*)d_in[2];
    const float* mlp_w = (const float*)d_in[3];
    const float* mlp_b = (const float*)d_in[4];
    const float* ew1   = (const float*)d_in[5];
    const float* ew2   = (const float*)d_in[6];
    const float* bn_g  = (const float*)d_in[7];
    const float* bn_b  = (const float*)d_in[8];
    float* out = (float*)d_out;

    const int* row = ei;
    const int* col = ei + N_EDGES;

    char* ws = (char*)d_ws;
    size_t off = 0;
    auto carve = [&](size_t bytes) {
        char* p = ws + off;
        off += (bytes + 255) & ~(size_t)255;
        return p;
    };
    float* deg  = (float*)carve((size_t)N_NODES * 4);                 // becomes dinv in place
    float* nrm  = (float*)carve((size_t)N_EDGES * 4);
    float* agg  = (float*)carve((size_t)N_NODES * HIDDEN * 4);
    float* bufA = (float*)carve((size_t)N_NODES * HIDDEN * 4);
    float* bufB = (float*)carve((size_t)N_NODES * HIDDEN * 4);
    float* sums = (float*)carve(2 * HIDDEN * 4);
    float* ss   = (float*)carve(2 * HIDDEN * 4);
    (void)ws_size; (void)in_sizes; (void)n_in; (void)out_size;

    // gcn_norm (once per call — deterministic, recomputed every launch)
    (void)hipMemsetAsync(deg, 0, (size_t)N_NODES * 4, stream);
    deg_kernel <<<(N_EDGES + 255) / 256, 256, 0, stream>>>(col, deg);
    dinv_kernel<<<(N_NODES + 255) / 256, 256, 0, stream>>>(deg);
    norm_kernel<<<(N_EDGES + 255) / 256, 256, 0, stream>>>(row, col, deg, nrm);

    const float* hin = x;
    for (int l = 0; l < N_LAYERS; ++l) {
        float* hout = (l == N_LAYERS - 1) ? out : ((l & 1) ? bufB : bufA);

        (void)hipMemsetAsync(agg, 0, (size_t)N_NODES * HIDDEN * 4, stream);
        msg_scatter_kernel<<<N_EDGES, HIDDEN, 0, stream>>>(
            hin, ea, row, col, nrm,
            ew1 + (size_t)l * (HIDDEN / 2),
            ew2 + (size_t)l * 3 * (HIDDEN / 2), agg);

        dim3 g((N_NODES + 63) / 64, HIDDEN / 64);
        gemm_kernel<<<g, 128, 0, stream>>>(
            hin, agg,
            mlp_w + (size_t)l * KDIM * HIDDEN,
            mlp_b + (size_t)l * HIDDEN, hout);

        (void)hipMemsetAsync(sums, 0, 2 * HIDDEN * 4, stream);
        bn_stats_kernel   <<<(N_NODES + 63) / 64, HIDDEN, 0, stream>>>(hout, sums);
        bn_finalize_kernel<<<1, HIDDEN, 0, stream>>>(sums, bn_g + (size_t)l * HIDDEN,
                                                     bn_b + (size_t)l * HIDDEN, ss);
        bn_apply_kernel   <<<((size_t)N_NODES * HIDDEN) / 256, 256, 0, stream>>>(
            ss, hout, (l != N_LAYERS - 1) ? 1 : 0);

        hin = hout;
    }
}